// Bce_iou_loss_19447611916794
// MI455X (gfx1250) — compile-verified
//
#include <hip/hip_runtime.h>
#include <hip/hip_bf16.h>
#include <stdint.h>

// Problem geometry
#define Hd    512
#define Wd    512
#define Bn    64
#define PADR  15          // (31-1)/2
#define YBLK  128         // rows per workgroup block
#define INV_KK (1.0f / 961.0f)

// gfx1250 async global->LDS path (guarded: falls back to plain loads if the
// toolchain doesn't declare the builtins; host pass always takes fallback).
#if defined(__gfx1250__) && \
    __has_builtin(__builtin_amdgcn_global_load_async_to_lds_b32) && \
    __has_builtin(__builtin_amdgcn_s_wait_asynccnt)
#define USE_ASYNC_LDS 1
#else
#define USE_ASYNC_LDS 0
#endif

#if USE_ASYNC_LDS
typedef __attribute__((address_space(1))) int gint_t;   // global int*
typedef __attribute__((address_space(3))) int lint_t;   // LDS int*
__device__ __forceinline__ void async_ld_b32(const float* g, float* l) {
  // global generic addr == AS1 addr; LDS generic addr low 32 bits == LDS offset
  __builtin_amdgcn_global_load_async_to_lds_b32(
      (gint_t*)(uintptr_t)g, (lint_t*)(uint32_t)(uintptr_t)l, 0, 0);
}
#endif

// One wave (32 lanes) per workgroup. Strip of 32 columns, block of 128 rows.
// grid = (16 strips * 4 y-blocks, 64 images)
__global__ __launch_bounds__(32)
void bce_iou_main(const float* __restrict__ pred,
                  const float* __restrict__ mask,
                  float* __restrict__ partials) {
  const int lane = threadIdx.x;            // 0..31 (wave32)
  const int b    = blockIdx.y;             // image
  const int wg   = blockIdx.x;             // 0..63
  const int sx   = wg >> 2;                // strip index (0..15)
  const int yb   = (wg & 3) * YBLK;        // first output row of this block
  const int x0   = sx * 32;
  const int x    = x0 + lane;              // this lane's output column

  const float* mB = mask + (size_t)b * (Hd * Wd);
  const float* pB = pred + (size_t)b * (Hd * Wd);

  __shared__ float ring[32][32];           // 31-row history of hrow, per column
#if USE_ASYNC_LDS
  __shared__ float seg[2][64];             // double-buffered halo row segment
#endif
#pragma unroll
  for (int s = 0; s < 32; ++s) ring[s][lane] = 0.f;

  // Halo segment covers columns [x0-15, x0+46]; lane owns segment slots 2l, 2l+1.
  const int  c0 = x0 - PADR + 2 * lane;
  const int  c1 = c0 + 1;
  const bool v0 = (c0 >= 0) & (c0 < Wd);
  const bool v1 = (c1 >= 0) & (c1 < Wd);

  const int rstart = yb - PADR;            // first ingested row (may be < 0)
  const int rlast  = yb + YBLK - 1 + PADR; // last ingested row (may be >= Hd)

  float vs = 0.f, sw = 0.f, swb = 0.f, swi = 0.f;

#if USE_ASYNC_LDS
  // Pre-issue row rstart into buffer 0 (zero-pad invalid lanes via ds_store).
  if (rstart >= 0 && rstart < Hd) {
    const float* row = mB + (size_t)rstart * Wd;
    if (v0) async_ld_b32(row + c0, &seg[0][2 * lane]); else seg[0][2 * lane] = 0.f;
    if (v1) async_ld_b32(row + c1, &seg[0][2 * lane + 1]); else seg[0][2 * lane + 1] = 0.f;
  }
  int dbuf = 0;
#endif

  for (int r = rstart; r <= rlast; ++r) {
    const bool haveRow = (r >= 0) & (r < Hd);   // wave-uniform
    float a0 = 0.f, a1 = 0.f;

#if USE_ASYNC_LDS
    // Software pipeline: issue row r+1 into the other buffer, then wait for row r.
    bool issuedNext = false;
    {
      const int rn = r + 1;
      if (rn <= rlast && rn >= 0 && rn < Hd) {
        const float* row = mB + (size_t)rn * Wd;
        float* sb = &seg[dbuf ^ 1][0];
        if (v0) async_ld_b32(row + c0, sb + 2 * lane); else sb[2 * lane] = 0.f;
        if (v1) async_ld_b32(row + c1, sb + 2 * lane + 1); else sb[2 * lane + 1] = 0.f;
        issuedNext = true;
      }
    }
    if (haveRow) {
      if (issuedNext) __builtin_amdgcn_s_wait_asynccnt(2);  // row r done (in-order)
      else            __builtin_amdgcn_s_wait_asynccnt(0);
      asm volatile("" ::: "memory");
      a0 = seg[dbuf][2 * lane];
      a1 = seg[dbuf][2 * lane + 1];
    }
    dbuf ^= 1;
#else
    if (haveRow) {
      const float* row = mB + (size_t)r * Wd;
      a0 = v0 ? row[c0] : 0.f;
      a1 = v1 ? row[c1] : 0.f;
      if (v0 && r + 8 < Hd) __builtin_prefetch(row + 8 * Wd + c0, 0, 1);
    }
#endif

    // Horizontal 31-tap box sum of the ingested row via wave prefix scan.
    float hrow = 0.f;
    if (haveRow) {
      float T = a0 + a1;                         // per-lane pair sum
#pragma unroll
      for (int d = 1; d < 32; d <<= 1) {         // inclusive scan over pairs
        float t = __shfl_up(T, d);
        if (lane >= d) T += t;
      }
      // hrow(x0+l) = P_incl(seg l+30) - P_incl(seg l-1), zero-padded.
      const int m1   = (lane + 30) >> 1;         // 15..30
      float Tm1  = __shfl(T, m1);
      float Tm1m = __shfl(T, m1 - 1);
      float A0m1 = __shfl(a0, m1);
      const float P1 = (lane & 1) ? Tm1 : (Tm1m + A0m1);

      const int m0  = (lane - 1) >> 1;
      const int m0c = m0 < 0 ? 0 : m0;
      float Tm0  = __shfl(T, m0c);
      float Tm0m = __shfl(T, m0c > 0 ? m0c - 1 : 0);
      float A0m0 = __shfl(a0, m0c);
      const float P0 = (lane == 0) ? 0.f
                     : ((lane & 1) ? ((m0 > 0 ? Tm0m : 0.f) + A0m0) : Tm0);
      hrow = P1 - P0;
    }

    // Vertical running 31-tap sum via 32-deep LDS ring (per column, no conflicts).
    const int slotOut = (r - 31) & 31;
    const float old = ring[slotOut][lane];
    vs += hrow - old;
    ring[r & 31][lane] = hrow;

    // Emit loss terms once the window for y = r-15 is complete.
    const int y = r - PADR;
    if (y >= yb) {                                // y <= yb+127 by loop bound
      const size_t off = (size_t)y * Wd + x;
      const float mv = mB[off];
      const float pv = pB[off];
      if (y + 8 < Hd) __builtin_prefetch(pB + off + 8 * Wd, 0, 1);

      const float avg = vs * INV_KK;
      const float w   = 1.f + 5.f * fabsf(avg - mv);
      const float bce = fmaxf(pv, 0.f) - pv * mv + log1pf(expf(-fabsf(pv)));
      const float sg  = 1.f / (1.f + expf(-pv));
      const float inter = sg * mv;
      const float uni   = sg + mv;
      const float iou = 1.f - (inter + 1.f) / (uni - inter + 1.f);
      sw  += w;
      swb += w * bce;
      swi += w * iou;
    }
  }

  // Deterministic wave reduction, fixed partial slot per workgroup.
#pragma unroll
  for (int d = 16; d >= 1; d >>= 1) {
    sw  += __shfl_down(sw,  d);
    swb += __shfl_down(swb, d);
    swi += __shfl_down(swi, d);
  }
  if (lane == 0) {
    float* p = partials + (size_t)(b * 64 + wg) * 3;
    p[0] = sw; p[1] = swb; p[2] = swi;
  }
}

// Single wave: fixed-order per-image sums, then mean over batch.
__global__ __launch_bounds__(32)
void bce_iou_final(const float* __restrict__ partials, float* __restrict__ out) {
  const int lane = threadIdx.x;
  float acc = 0.f;
#pragma unroll
  for (int k = 0; k < 2; ++k) {
    const int b = lane + 32 * k;                 // 2 images per lane
    float w = 0.f, wb = 0.f, wi = 0.f;
    for (int i = 0; i < 64; ++i) {
      const float* p = partials + (size_t)(b * 64 + i) * 3;
      w += p[0]; wb += p[1]; wi += p[2];
    }
    acc += wb / w + wi / w;
  }
#pragma unroll
  for (int d = 16; d >= 1; d >>= 1) acc += __shfl_down(acc, d);
  if (lane == 0) out[0] = acc * (1.0f / 64.0f);
}

extern "C" void kernel_launch(void* const* d_in, const int* in_sizes, int n_in,
                              void* d_out, int out_size, void* d_ws, size_t ws_size,
                              hipStream_t stream) {
  (void)in_sizes; (void)n_in; (void)out_size; (void)ws_size;
  const float* pred = (const float*)d_in[0];   // (64,1,512,512) f32
  const float* mask = (const float*)d_in[1];   // (64,1,512,512) f32
  float* partials = (float*)d_ws;              // 4096 * 3 floats = 48 KB

  dim3 grid(64, Bn);                           // 16 strips * 4 y-blocks, 64 images
  bce_iou_main<<<grid, 32, 0, stream>>>(pred, mask, partials);
  bce_iou_final<<<1, 32, 0, stream>>>(partials, (float*)d_out);
}